// HamiltonianEncoder_38336878084597
// MI455X (gfx1250) — compile-verified
//
#include <hip/hip_runtime.h>
#include <cmath>

// ---------------------------------------------------------------------------
// HamiltonianEncoder for MI455X (gfx1250, wave32, WMMA).
//
// Shapes: V=32000, D=512, S=1024, H=2048, B=32, R=B*S=32768.
// q = d_out[0..R*D), p = d_out[R*D..2*R*D).
// d_ws: mask[R], mass[S], W1T[D*H]  (~4.3 MB total, W1T is L2-resident).
//
// Leapfrog with merged half-kicks:
//   kick(0.5h); 4x{ drift; kick(h) }; drift; kick(0.5h)
// kick: p = (p + coef * -(mask * g(q))) * mask   (fused GEMM epilogue)
//   g[m,d] = sum_h sech^2(q[m,:]·W1[h,:] + b1[h]) * w2[h] * W1[h,d]
//   sech^2(x) = 4e/(1+e)^2, e = exp(-2|x|)   (branch-free, no tanh slow path)
// drift: q += h * (p / mass) * mask
// ---------------------------------------------------------------------------

typedef __attribute__((ext_vector_type(2))) float v2f;
typedef __attribute__((ext_vector_type(8))) float v8f;

#define DIM_D 512
#define DIM_H 2048
#define HCHUNK 256
#define MT 16
#define TPB 128   // 4 waves

__device__ __forceinline__ v8f wmma_f32_4(v2f a, v2f b, v8f c) {
  return __builtin_amdgcn_wmma_f32_16x16x4_f32(
      /*neg_a=*/false, a, /*neg_b=*/false, b,
      /*c_mod=*/(short)0, c, /*reuse_a=*/false, /*reuse_b=*/false);
}

// --------------------------- embed + momentum init -------------------------
__global__ __launch_bounds__(256)
void embed_kernel(const float* __restrict__ table,
                  const float* __restrict__ log_masses,
                  const int* __restrict__ ids,
                  float* __restrict__ q, float* __restrict__ p,
                  float* __restrict__ mask, int S) {
  const int r = blockIdx.x;              // row in [0, B*S)
  const int s = r % S;
  const int id = ids[r];
  const int idp = (s == 0) ? id : ids[r - 1];
  const float mk = (id != 0) ? 1.0f : 0.0f;
  const float mkp = (idp != 0) ? 1.0f : 0.0f;
  const float mass = __expf(log_masses[s]);
  if (threadIdx.x == 0) mask[r] = mk;
  for (int d = threadIdx.x; d < DIM_D; d += 256) {
    float qv = mk  * table[(size_t)id  * DIM_D + d];
    float qp = mkp * table[(size_t)idp * DIM_D + d];
    if (s == 0) qp = qv;                  // replicate-pad -> vel = 0
    q[(size_t)r * DIM_D + d] = qv;
    p[(size_t)r * DIM_D + d] = mass * (qv - qp) * mk;   // DT_MOMENTUM = 1
  }
}

__global__ __launch_bounds__(256)
void mass_kernel(const float* __restrict__ log_masses,
                 float* __restrict__ mass, int S) {
  int s = blockIdx.x * 256 + threadIdx.x;
  if (s < S) mass[s] = __expf(log_masses[s]);
}

// ------------------------ W1 transpose (once per launch) -------------------
// W1T[d][h] = W1[h][d]; makes GEMM2's B operand contiguous along k per lane.
__global__ __launch_bounds__(256)
void transpose_kernel(const float* __restrict__ W1, float* __restrict__ W1T) {
  __shared__ float tile[32][33];
  const int bx = blockIdx.x;             // D / 32 tiles
  const int by = blockIdx.y;             // H / 32 tiles
  const int tx = threadIdx.x & 31;
  const int ty = threadIdx.x >> 5;       // 0..7
  #pragma unroll
  for (int i = 0; i < 32; i += 8)
    tile[ty + i][tx] = W1[(size_t)(by * 32 + ty + i) * DIM_D + bx * 32 + tx];
  __syncthreads();
  #pragma unroll
  for (int i = 0; i < 32; i += 8)
    W1T[(size_t)(bx * 32 + ty + i) * DIM_H + by * 32 + tx] = tile[tx][ty + i];
}

// --------------------------------- drift -----------------------------------
__global__ __launch_bounds__(256)
void drift_kernel(float* __restrict__ q, const float* __restrict__ p,
                  const float* __restrict__ mask, const float* __restrict__ mass,
                  int S, int n, float h) {
  int i = blockIdx.x * 256 + threadIdx.x;
  if (i >= n) return;
  int r = i >> 9;                 // / DIM_D
  int s = r % S;
  q[i] += h * (p[i] / mass[s]) * mask[r];
}

// ------------------------------ fused kick ---------------------------------
__global__ __launch_bounds__(TPB)
void kick_kernel(const float* __restrict__ q,
                 float* __restrict__ p,
                 const float* __restrict__ W1,   // [H, D] row-major (GEMM1 B)
                 const float* __restrict__ W1T,  // [D, H] row-major (GEMM2 B)
                 const float* __restrict__ b1,   // [H]
                 const float* __restrict__ w2,   // [H]
                 const float* __restrict__ mask, // [R]
                 float coef) {
  __shared__ float sQ[MT][DIM_D];   // 32 KB : q row tile
  __shared__ float sA[MT][HCHUNK];  // 16 KB : sech^2 * w2 chunk (GEMM2 A)

  const int rowBase = blockIdx.x * MT;
  const int tid  = threadIdx.x;
  const int wave = tid >> 5;
  const int lane = tid & 31;
  const int half = lane >> 4;       // 0: lanes 0-15, 1: lanes 16-31
  const int l16  = lane & 15;

  // stage q tile
  for (int i = tid; i < MT * DIM_D; i += TPB) {
    int m = i >> 9, k = i & (DIM_D - 1);
    sQ[m][k] = q[(size_t)(rowBase + m) * DIM_D + k];
  }
  __syncthreads();

  // g accumulators: this wave owns d in [wave*128, wave*128+128) -> 8 tiles
  v8f acc[8] = {};

  // per-lane base for GEMM2 B loads (row d0+l16 of W1T, contiguous along k)
  const float* b2base = &W1T[(size_t)(wave * 128 + l16) * DIM_H + 2 * half];

  for (int hc = 0; hc < DIM_H; hc += HCHUNK) {
    // ---- GEMM1: activation chunk, this wave computes 64 h-columns ----
    for (int nt = 0; nt < 4; ++nt) {
      const int h0 = hc + wave * 64 + nt * 16;
      v8f c = {};
      // A (16x4 f32): lanes<16 -> M=l16, K={k0,k0+1}; lanes>=16 -> K={k0+2,k0+3}
      // B[k][n] = W1[h0+n][k]  (contiguous float2 per lane along k)
      const float* brow = &W1[(size_t)(h0 + l16) * DIM_D + 2 * half];
      #pragma unroll 8
      for (int k0 = 0; k0 < DIM_D; k0 += 4) {
        const float* ap = &sQ[l16][k0 + 2 * half];
        v2f a; a.x = ap[0]; a.y = ap[1];
        v2f b; b.x = brow[k0]; b.y = brow[k0 + 1];
        c = wmma_f32_4(a, b, c);
      }
      // C layout: element (v,lane) -> M = v + 8*half, N = l16 (col h0+l16)
      // a = (1 - tanh^2(x)) * w2 = 4*e/(1+e)^2 * w2,  e = exp(-2|x|)
      const float bb  = b1[h0 + l16];
      const float ww4 = 4.0f * w2[h0 + l16];
      #pragma unroll
      for (int v = 0; v < 8; ++v) {
        float x = c[v] + bb;
        float e = __expf(-2.0f * __builtin_fabsf(x));
        float rv = __builtin_amdgcn_rcpf(1.0f + e);
        sA[v + 8 * half][wave * 64 + nt * 16 + l16] = ww4 * e * rv * rv;
      }
    }
    __syncthreads();

    // ---- GEMM2: acc[nt] += a_chunk @ W1[hc:hc+HCHUNK, :]  (B from W1T) ----
    #pragma unroll 4
    for (int k0 = 0; k0 < HCHUNK; k0 += 4) {
      const float* ap = &sA[l16][k0 + 2 * half];
      v2f a; a.x = ap[0]; a.y = ap[1];
      #pragma unroll
      for (int nt = 0; nt < 8; ++nt) {
        // B[k][n] = W1T[d0+n][hc+k] : contiguous float2 per lane along k
        const float* bp = b2base + (size_t)nt * 16 * DIM_H + hc + k0;
        v2f b; b.x = bp[0]; b.y = bp[1];
        acc[nt] = wmma_f32_4(a, b, acc[nt]);
      }
    }
    __syncthreads();   // before next chunk overwrites sA
  }

  // ---- epilogue: p = (p + coef * -(mask*g)) * mask ----
  #pragma unroll
  for (int nt = 0; nt < 8; ++nt) {
    const int d0 = wave * 128 + nt * 16;
    #pragma unroll
    for (int v = 0; v < 8; ++v) {
      const int m = v + 8 * half;
      const int r = rowBase + m;
      const float mk = mask[r];
      const size_t idx = (size_t)r * DIM_D + d0 + l16;
      const float f = -(mk * acc[nt][v]);
      p[idx] = (p[idx] + coef * f) * mk;
    }
  }
}

// ------------------------------- launcher ----------------------------------
extern "C" void kernel_launch(void* const* d_in, const int* in_sizes, int n_in,
                              void* d_out, int out_size, void* d_ws, size_t ws_size,
                              hipStream_t stream) {
  const float* table      = (const float*)d_in[0];
  const float* log_masses = (const float*)d_in[1];
  const float* W1         = (const float*)d_in[2];
  const float* b1         = (const float*)d_in[3];
  const float* W2         = (const float*)d_in[4];   // [1,H] -> w2 row
  // d_in[5] = b2: constant offset, zero force gradient -> unused
  const int*   ids        = (const int*)d_in[6];

  const int S = in_sizes[1];              // 1024
  const int R = in_sizes[6];              // B*S = 32768
  const int N = R * DIM_D;                // elements per field

  float* qbuf = (float*)d_out;            // [R, D]
  float* pbuf = (float*)d_out + (size_t)N;
  float* mask = (float*)d_ws;             // [R]
  float* mass = mask + R;                 // [S]
  float* W1T  = mass + S;                 // [D, H] = 4 MB

  const float h = 0.1f;                   // H_STEP

  mass_kernel<<<(S + 255) / 256, 256, 0, stream>>>(log_masses, mass, S);
  transpose_kernel<<<dim3(DIM_D / 32, DIM_H / 32), 256, 0, stream>>>(W1, W1T);
  embed_kernel<<<R, 256, 0, stream>>>(table, log_masses, ids, qbuf, pbuf, mask, S);

  const int kickBlocks  = R / MT;
  const int driftBlocks = (N + 255) / 256;

  // kick(h/2); 4x{ drift; kick(h) }; drift; kick(h/2)   == 5 leapfrog steps
  kick_kernel<<<kickBlocks, TPB, 0, stream>>>(qbuf, pbuf, W1, W1T, b1, W2, mask, 0.5f * h);
  for (int s = 0; s < 4; ++s) {
    drift_kernel<<<driftBlocks, 256, 0, stream>>>(qbuf, pbuf, mask, mass, S, N, h);
    kick_kernel<<<kickBlocks, TPB, 0, stream>>>(qbuf, pbuf, W1, W1T, b1, W2, mask, h);
  }
  drift_kernel<<<driftBlocks, 256, 0, stream>>>(qbuf, pbuf, mask, mass, S, N, h);
  kick_kernel<<<kickBlocks, TPB, 0, stream>>>(qbuf, pbuf, W1, W1T, b1, W2, mask, 0.5f * h);
}